// Context_Attention_18803366822321
// MI455X (gfx1250) — compile-verified
//
#include <hip/hip_runtime.h>
#include <math.h>

// Problem constants (from reference)
#define Hn 512
#define Bn 64
#define Pn 32
#define Rn 4
#define Wn 64
#define HP 516                     // padded LDS row stride (floats): bank = (4m+h)%64, conflict-free
#define ALPHA_OFF (Rn * Bn * Hn)   // resp_c occupies first R*B*H floats of d_out

typedef __attribute__((ext_vector_type(2))) float v2f;
typedef __attribute__((ext_vector_type(8))) float v8f;

__device__ __forceinline__ v8f wmma4(v2f a, v2f b, v8f c) {
  // D(16x16,f32) += A(16x4,f32) * B(4x16,f32)
  return __builtin_amdgcn_wmma_f32_16x16x4_f32(false, a, false, b, (short)0, c, false, false);
}

__device__ __forceinline__ float fast_tanh(float x) {
#if __has_builtin(__builtin_amdgcn_tanhf)
  return __builtin_amdgcn_tanhf(x);   // gfx1250 V_TANH_F32
#else
  return tanhf(x);
#endif
}

// Async global->LDS copy of one 16B chunk per lane (CDNA5, ASYNCcnt-tracked).
// Generic shared pointers carry the LDS byte offset in their low 32 bits
// (flat aperture mapping truncates to addr[31:0]), which is what VDST needs.
__device__ __forceinline__ void async_g2l_b128(const float* gsrc, void* ldst) {
  uint32_t ldsoff = (uint32_t)(uintptr_t)ldst;
  asm volatile("global_load_async_to_lds_b128 %0, %1, off"
               :: "v"(ldsoff), "v"(gsrc)
               : "memory");
}
__device__ __forceinline__ void wait_async0() {
  asm volatile("s_wait_asynccnt 0" ::: "memory");
}

// ---------------------------------------------------------------------------
// Kernel 1: Wctx[b*P+p, k] = sum_h context_output[p,b,h] * W_c[h,k]
// Grid: 128 blocks (16-row tiles of the 2048-row A), 256 threads (8 wave32).
// ---------------------------------------------------------------------------
__global__ void __launch_bounds__(256)
wctx_kernel(const float* __restrict__ ctx, const float* __restrict__ Wc,
            float* __restrict__ wctx) {
  const int tid  = threadIdx.x;
  const int lane = tid & 31;
  const int wave = tid >> 5;
  const int rt   = blockIdx.x;
  const int m    = lane & 15;
  const int n    = lane & 15;
  const int kb   = (lane >> 4) << 1;          // 0 (lanes 0-15) or 2 (lanes 16-31)
  const int moff = (lane < 16) ? 0 : 8;

  const int u = rt * 16 + m;                  // row index = b*32 + p
  const int b = u >> 5, p = u & 31;
  const float* __restrict__ arow = ctx + ((size_t)p * Bn + b) * Hn;

  for (int ct = wave; ct < Hn / 16; ct += 8) {
    const int n0 = ct * 16;
    v8f acc = {};
    for (int h = 0; h < Hn; h += 4) {
      v2f a  = { arow[h + kb], arow[h + kb + 1] };
      v2f bb = { Wc[(size_t)(h + kb) * Hn + n0 + n],
                 Wc[(size_t)(h + kb + 1) * Hn + n0 + n] };
      acc = wmma4(a, bb, acc);
    }
    for (int j = 0; j < 8; ++j)
      wctx[(size_t)(rt * 16 + moff + j) * Hn + n0 + n] = acc[j];
  }
}

// ---------------------------------------------------------------------------
// Kernel 2: full sequential scan. 16 blocks; block owns rows g = blk*16..+15
// where g = r*64 + b. State atten[16][H] lives in LDS across all 64 steps.
// resp_t tiles are double-buffered via async global->LDS prefetch.
// ---------------------------------------------------------------------------
__global__ void __launch_bounds__(256)
scan_kernel(const float* __restrict__ ctx,        // (P,B,H)
            const float* __restrict__ resp_out,   // (R,W,B,H)
            const float* __restrict__ resp_hid,   // (R,B,H)
            const float* __restrict__ Wt,         // (H,H)
            const float* __restrict__ Wa,         // (H,H)
            const float* __restrict__ Walpha,     // (H)
            const float* __restrict__ lin_w,      // (H,H), used transposed
            const float* __restrict__ lin_p,      // (H,H), used transposed
            const float* __restrict__ lin_x,      // (H,H), used transposed
            const float* __restrict__ wctx,       // (B*P,H) precomputed
            float* __restrict__ out) {
  extern __shared__ float smem[];
  float* s_atten = smem;              // 16 x HP  (recurrent state)
  float* s_w     = smem + 16 * HP;    // 16 x HP  (w, then tanh(lin))
  float* s_x     = smem + 32 * HP;    // 2 x 16 x HP (double-buffered resp_t)
  float* s_alpha = smem + 64 * HP;    // 16 x 32  (logits -> alpha)

  const int tid  = threadIdx.x;
  const int lane = tid & 31;
  const int wave = tid >> 5;
  const int blk  = blockIdx.x;
  const int m    = lane & 15;
  const int n    = lane & 15;
  const int kb   = (lane >> 4) << 1;
  const int moff = (lane < 16) ? 0 : 8;

  // per-lane register cache of w_alpha for the fixed h = lane + 32*it pattern
  float wal[16];
  for (int it = 0; it < 16; ++it) wal[it] = Walpha[lane + (it << 5)];

  for (int i = tid; i < 16 * HP; i += 256) s_atten[i] = 0.f;

  // prime: async prefetch resp_t tile for t = 0 into buffer 0
  for (int j = tid; j < 16 * 128; j += 256) {      // 16 rows x 128 b128-chunks
    const int row = j >> 7, chunk = j & 127;
    const int g = blk * 16 + row;
    const int r = g >> 6, b = g & 63;
    const float* gsrc = resp_out + (((size_t)r * Wn + 0) * Bn + b) * Hn + chunk * 4;
    async_g2l_b128(gsrc, s_x + row * HP + chunk * 4);
  }
  wait_async0();
  __syncthreads();

  v8f lin_acc[4];

  for (int t = 0; t < Wn; ++t) {
    float* s_xc = s_x + (t & 1) * 16 * HP;          // current buffer
    // ---- async prefetch next step's resp_t tile into the other buffer
    if (t + 1 < Wn) {
      float* s_xn = s_x + ((t + 1) & 1) * 16 * HP;
      for (int j = tid; j < 16 * 128; j += 256) {
        const int row = j >> 7, chunk = j & 127;
        const int g = blk * 16 + row;
        const int r = g >> 6, b = g & 63;
        const float* gsrc =
            resp_out + (((size_t)r * Wn + (t + 1)) * Bn + b) * Hn + chunk * 4;
        async_g2l_b128(gsrc, s_xn + row * HP + chunk * 4);
      }
    }

    // ---- phase 1: fused GEMMs: w = x@Wt + atten@Wa (to LDS); lin = atten@lin_w^T (regs)
    {
      const float* __restrict__ xrow = s_xc + m * HP;
      const float* __restrict__ srow = s_atten + m * HP;
      int cti = 0;
      for (int ct = wave; ct < 32; ct += 8, ++cti) {
        const int n0 = ct * 16;
        v8f accw = {};
        v8f accl = {};
        for (int h = 0; h < Hn; h += 4) {
          v2f ax = { xrow[h + kb], xrow[h + kb + 1] };
          v2f bt = { Wt[(size_t)(h + kb) * Hn + n0 + n],
                     Wt[(size_t)(h + kb + 1) * Hn + n0 + n] };
          accw = wmma4(ax, bt, accw);
          v2f aa = { srow[h + kb], srow[h + kb + 1] };
          v2f ba = { Wa[(size_t)(h + kb) * Hn + n0 + n],
                     Wa[(size_t)(h + kb + 1) * Hn + n0 + n] };
          accw = wmma4(aa, ba, accw);
          v2f bl = { lin_w[(size_t)(n0 + n) * Hn + h + kb],
                     lin_w[(size_t)(n0 + n) * Hn + h + kb + 1] };
          accl = wmma4(aa, bl, accl);
        }
        for (int j = 0; j < 8; ++j)
          s_w[(moff + j) * HP + n0 + n] = accw[j];
        lin_acc[cti] = accl;
      }
    }
    __syncthreads();

    // ---- phase 2: logits[row,p] = sum_h tanh(Wctx[b,p,h] + w[row,h]) * walpha[h]
    // wave-cooperative: lanes split h (coalesced Wctx reads), shuffle-reduce.
    for (int pr = wave; pr < 16 * Pn; pr += 8) {
      const int row = pr >> 5, p = pr & 31;
      const int g = blk * 16 + row;
      const int b = g & 63;
      const float* __restrict__ wcrow = wctx + ((size_t)b * Pn + p) * Hn + lane;
      const float* __restrict__ wrow  = s_w + row * HP + lane;
      float acc = 0.f;
      for (int it = 0; it < 16; ++it)
        acc += fast_tanh(wcrow[it << 5] + wrow[it << 5]) * wal[it];
      acc += __shfl_xor(acc, 16, 32);
      acc += __shfl_xor(acc, 8, 32);
      acc += __shfl_xor(acc, 4, 32);
      acc += __shfl_xor(acc, 2, 32);
      acc += __shfl_xor(acc, 1, 32);
      if (lane == 0) s_alpha[pr] = acc;
    }
    __syncthreads();

    // ---- phase 3: overwrite s_w with tanh(lin); softmax over P in s_alpha
    {
      int cti = 0;
      for (int ct = wave; ct < 32; ct += 8, ++cti) {
        const int n0 = ct * 16;
        for (int j = 0; j < 8; ++j)
          s_w[(moff + j) * HP + n0 + n] = fast_tanh(lin_acc[cti][j]);
      }
    }
    if (tid < 16) {
      float* lrow = s_alpha + tid * Pn;
      float mx = lrow[0];
      for (int p = 1; p < Pn; ++p) mx = fmaxf(mx, lrow[p]);
      float sum = 0.f;
      for (int p = 0; p < Pn; ++p) { float e = __expf(lrow[p] - mx); lrow[p] = e; sum += e; }
      const float inv = 1.f / sum;
      for (int p = 0; p < Pn; ++p) lrow[p] *= inv;
    }
    __syncthreads();

    // ---- phase 4: emit alpha; atten = alpha@ctx + tanh(lin)
    for (int i = tid; i < 16 * Pn; i += 256) {
      const int row = i >> 5, p = i & 31;
      const int g = blk * 16 + row;
      const int r = g >> 6, b = g & 63;
      out[ALPHA_OFF + (((size_t)r * Wn + t) * Bn + b) * Pn + p] = s_alpha[row * Pn + p];
    }
    // wave-cooperative: wave handles 2 rows; lanes cover consecutive h (coalesced ctx reads)
    for (int row = wave * 2; row < wave * 2 + 2; ++row) {
      const int g = blk * 16 + row;
      const int b = g & 63;
      float acc[16];
      for (int it = 0; it < 16; ++it) acc[it] = 0.f;
      for (int p = 0; p < Pn; ++p) {
        const float a = s_alpha[row * Pn + p];
        const float* __restrict__ crow = ctx + ((size_t)p * Bn + b) * Hn + lane;
        for (int it = 0; it < 16; ++it)
          acc[it] += a * crow[it << 5];
      }
      for (int it = 0; it < 16; ++it) {
        const int h = lane + (it << 5);
        s_atten[row * HP + h] = acc[it] + s_w[row * HP + h];
      }
    }
    wait_async0();      // next-step resp_t tile resident before barrier
    __syncthreads();
  }

  // ---- epilogue: resp_c = tanh(atten@lin_p^T + resp_hidden@lin_x^T)
  {
    const int g = blk * 16 + m;
    const float* __restrict__ hrow = resp_hid + (size_t)g * Hn;
    const float* __restrict__ srow = s_atten + m * HP;
    for (int ct = wave; ct < 32; ct += 8) {
      const int n0 = ct * 16;
      v8f acc = {};
      for (int h = 0; h < Hn; h += 4) {
        v2f aa = { srow[h + kb], srow[h + kb + 1] };
        v2f bp = { lin_p[(size_t)(n0 + n) * Hn + h + kb],
                   lin_p[(size_t)(n0 + n) * Hn + h + kb + 1] };
        acc = wmma4(aa, bp, acc);
        v2f ah = { hrow[h + kb], hrow[h + kb + 1] };
        v2f bx = { lin_x[(size_t)(n0 + n) * Hn + h + kb],
                   lin_x[(size_t)(n0 + n) * Hn + h + kb + 1] };
        acc = wmma4(ah, bx, acc);
      }
      for (int j = 0; j < 8; ++j)
        out[(size_t)(blk * 16 + moff + j) * Hn + n0 + n] = fast_tanh(acc[j]);
    }
  }
}

// ---------------------------------------------------------------------------
extern "C" void kernel_launch(void* const* d_in, const int* in_sizes, int n_in,
                              void* d_out, int out_size, void* d_ws, size_t ws_size,
                              hipStream_t stream) {
  const float* context_output = (const float*)d_in[0];
  const float* resp_output   = (const float*)d_in[1];
  const float* resp_hidden   = (const float*)d_in[2];
  const float* W_c           = (const float*)d_in[3];
  const float* W_t           = (const float*)d_in[4];
  const float* W_a           = (const float*)d_in[5];
  const float* W_alpha       = (const float*)d_in[6];
  const float* lin_w         = (const float*)d_in[7];
  const float* lin_p_w       = (const float*)d_in[8];
  const float* lin_x_w       = (const float*)d_in[9];
  float* out  = (float*)d_out;
  float* wctx = (float*)d_ws;               // needs B*P*H*4 = 4 MB scratch

  wctx_kernel<<<dim3(Bn * Pn / 16), dim3(256), 0, stream>>>(context_output, W_c, wctx);

  const size_t shmem = (size_t)(64 * HP + 16 * Pn) * sizeof(float);  // ~131 KB
  (void)hipFuncSetAttribute((const void*)scan_kernel,
                            hipFuncAttributeMaxDynamicSharedMemorySize, (int)shmem);
  scan_kernel<<<dim3((Rn * Bn) / 16), dim3(256), shmem, stream>>>(
      context_output, resp_output, resp_hidden, W_t, W_a, W_alpha,
      lin_w, lin_p_w, lin_x_w, wctx, out);
}